// SAGE_26671746908236
// MI455X (gfx1250) — compile-verified
//
#include <hip/hip_runtime.h>

#define N_NODES 50000
#define N_EDGES 800000
#define NFEAT 64
#define NHID 64
#define NCLASS 16
#define NTILES (N_NODES / 16)      // 3125, exact
#define WPB 8                      // waves per block
#define HPAD 68                    // padded LDS stride for h staging (bank-conflict-free)

typedef __attribute__((ext_vector_type(2))) float v2f;
typedef __attribute__((ext_vector_type(8))) float v8f;

// ---------------------------------------------------------------- zero scratch
__global__ void SAGE_zero(float* __restrict__ p, int n) {
    int i = blockIdx.x * blockDim.x + threadIdx.x;
    int stride = gridDim.x * blockDim.x;
    for (; i < n; i += stride) p[i] = 0.0f;
}

// ------------------------------------------------- edge scatter: sum + degree
// 16 threads per edge; each thread gathers a float4 of x[src] and does 4
// hardware f32 atomics into the L2-resident accumulator (12.8 MB << 192 MB L2).
__global__ void SAGE_scatter(const float* __restrict__ x,
                             const int* __restrict__ src,
                             const int* __restrict__ dst,
                             float* __restrict__ nsum,
                             float* __restrict__ deg) {
    int tid = blockIdx.x * blockDim.x + threadIdx.x;
    int e = tid >> 4;
    int q = tid & 15;
    if (e >= N_EDGES) return;
    int s = src[e];
    int d = dst[e];
    const float4 xv = *(const float4*)(x + (long long)s * NFEAT + q * 4);
    float* sp = nsum + (long long)d * NFEAT + q * 4;
    unsafeAtomicAdd(sp + 0, xv.x);
    unsafeAtomicAdd(sp + 1, xv.y);
    unsafeAtomicAdd(sp + 2, xv.z);
    unsafeAtomicAdd(sp + 3, xv.w);
    if (q == 0) unsafeAtomicAdd(deg + d, 1.0f);
}

// --------------------------------------- fused dense phase, f32 WMMA (16x16x4)
// One wave per 16-node tile. Layouts per CDNA5 ISA 7.12.2:
//   A 16x4 f32 : lane L -> row m=L%16 ; .x = K(k0+2*half), .y = K(+1)
//   B  4x16 f32: lane L -> col n=L%16 ; .x = row k0+2*half, .y = row (+1)
//   C/D 16x16  : lane L -> col n=L%16 ; vgpr i -> row i + 8*half
__global__ __launch_bounds__(256) void SAGE_dense_wmma(
    const float* __restrict__ x,
    const float* __restrict__ nsum,
    const float* __restrict__ deg,
    const float* __restrict__ Wself,
    const float* __restrict__ Wneigh,
    const float* __restrict__ bsage,
    const float* __restrict__ Wfc,
    const float* __restrict__ bfc,
    float* __restrict__ out) {
    __shared__ float sW[2 * NFEAT * NHID + NHID * NCLASS];   // 9216 f = 36 KB
    __shared__ float sH[WPB][16][HPAD];                      // h staging, 34 KB

    // stage all weights in LDS once per block
    for (int i = threadIdx.x; i < NFEAT * NHID; i += blockDim.x) {
        sW[i] = Wself[i];
        sW[NFEAT * NHID + i] = Wneigh[i];
    }
    for (int i = threadIdx.x; i < NHID * NCLASS; i += blockDim.x)
        sW[2 * NFEAT * NHID + i] = Wfc[i];
    __syncthreads();

    const int lane = threadIdx.x & 31;
    const int wave = threadIdx.x >> 5;
    const int m    = lane & 15;   // A-row / B-C-D-column index
    const int half = lane >> 4;   // wave half select

    int tile = blockIdx.x * WPB + wave;
    if (tile >= NTILES) tile = NTILES - 1;   // keep EXEC all-ones; dup writes identical

    const int arow = tile * 16 + m;          // node this lane reads (A layout)
    const float inv = 1.0f / fmaxf(deg[arow], 1.0f);
    const float* __restrict__ xrow = x    + (long long)arow * NFEAT;
    const float* __restrict__ srow = nsum + (long long)arow * NFEAT;
    const float* __restrict__ sWn  = sW + NFEAT * NHID;
    const float* __restrict__ sWf  = sW + 2 * NFEAT * NHID;

    // hidden accumulators, bias pre-loaded (bias is per-column == per-lane-n)
    v8f acc[4];
#pragma unroll
    for (int t = 0; t < 4; ++t) {
        float b = bsage[t * 16 + m];
#pragma unroll
        for (int i = 0; i < 8; ++i) acc[t][i] = b;
    }

    // h = x@Wself + (nsum/deg)@Wneigh  (+bias), K=64 in steps of 4
    for (int k0 = 0; k0 < NFEAT; k0 += 4) {
        const int ka = k0 + 2 * half;
        v2f ax, ah;
        ax.x = xrow[ka];       ax.y = xrow[ka + 1];
        ah.x = srow[ka] * inv; ah.y = srow[ka + 1] * inv;
#pragma unroll
        for (int t = 0; t < 4; ++t) {
            const int n = t * 16 + m;
            v2f bs, bn;
            bs.x = sW [ka * NHID + n];  bs.y = sW [(ka + 1) * NHID + n];
            bn.x = sWn[ka * NHID + n];  bn.y = sWn[(ka + 1) * NHID + n];
            acc[t] = __builtin_amdgcn_wmma_f32_16x16x4_f32(
                false, ax, false, bs, (short)0, acc[t], false, false);
            acc[t] = __builtin_amdgcn_wmma_f32_16x16x4_f32(
                false, ah, false, bn, (short)0, acc[t], false, false);
        }
    }

    // ReLU and stage h tile to LDS (C/D layout -> memory row-major)
#pragma unroll
    for (int t = 0; t < 4; ++t)
#pragma unroll
        for (int i = 0; i < 8; ++i)
            sH[wave][i + 8 * half][t * 16 + m] = fmaxf(acc[t][i], 0.0f);
    __syncthreads();

    // out = h @ Wfc + bfc   (16x64 @ 64x16)
    v8f oacc;
    {
        float b = bfc[m];
#pragma unroll
        for (int i = 0; i < 8; ++i) oacc[i] = b;
    }
    for (int k0 = 0; k0 < NHID; k0 += 4) {
        const int ka = k0 + 2 * half;
        v2f a, b;
        a.x = sH[wave][m][ka];      a.y = sH[wave][m][ka + 1];
        b.x = sWf[ka * NCLASS + m]; b.y = sWf[(ka + 1) * NCLASS + m];
        oacc = __builtin_amdgcn_wmma_f32_16x16x4_f32(
            false, a, false, b, (short)0, oacc, false, false);
    }

    // store D (16x16) -> out rows tile*16 + i + 8*half, col m
#pragma unroll
    for (int i = 0; i < 8; ++i)
        out[(long long)(tile * 16 + i + 8 * half) * NCLASS + m] = oacc[i];
}

// ---------------------------------------------------------------------- launch
extern "C" void kernel_launch(void* const* d_in, const int* in_sizes, int n_in,
                              void* d_out, int out_size, void* d_ws, size_t ws_size,
                              hipStream_t stream) {
    (void)in_sizes; (void)n_in; (void)out_size; (void)ws_size;
    const float* x      = (const float*)d_in[0];
    const int*   src    = (const int*)d_in[1];
    const int*   dst    = (const int*)d_in[2];
    const float* Wself  = (const float*)d_in[3];
    const float* Wneigh = (const float*)d_in[4];
    const float* bsage  = (const float*)d_in[5];
    const float* Wfc    = (const float*)d_in[6];
    const float* bfc    = (const float*)d_in[7];
    float* out  = (float*)d_out;
    float* nsum = (float*)d_ws;                       // [N_NODES * NFEAT]
    float* deg  = nsum + (size_t)N_NODES * NFEAT;     // [N_NODES]

    const int nzero = N_NODES * NFEAT + N_NODES;
    SAGE_zero<<<2048, 256, 0, stream>>>(nsum, nzero);

    const int sthreads = N_EDGES * 16;                // 12.8M
    SAGE_scatter<<<sthreads / 256, 256, 0, stream>>>(x, src, dst, nsum, deg);

    SAGE_dense_wmma<<<(NTILES + WPB - 1) / WPB, 256, 0, stream>>>(
        x, nsum, deg, Wself, Wneigh, bsage, Wfc, bfc, out);
}